// Fusedmax_74019466379623
// MI455X (gfx1250) — compile-verified
//
#include <hip/hip_runtime.h>
#include <stdint.h>

#define NCOLS 512
#define LAMBDA 1.0f

// One row per 32-thread (single-wave) workgroup.
//  Phase 1: async global->LDS copy of the row (CDNA5 ASYNCcnt path).
//  Phase 2: lane 0 runs Condat's TV-1D prox serially out of LDS (exact port
//           of the reference's while-loop body).
//  Phase 3: all lanes run Michelot's simplex projection (== sparsemax) with
//           wave32 shuffle reductions; coalesced store.
__global__ __launch_bounds__(32) void fusedmax_tv_sparsemax(
    const float* __restrict__ xin, float* __restrict__ out, int nrows) {
  __shared__ float sy[NCOLS];   // TV input
  __shared__ float sx[NCOLS];   // TV output
  const int row = blockIdx.x;
  if (row >= nrows) return;
  const int lane = (int)threadIdx.x;
  const float* __restrict__ src = xin + (size_t)row * NCOLS;

#if defined(__gfx1250__)
  // 16 bytes/lane/op: 4 ops move the whole 2KB row into LDS asynchronously.
  #pragma unroll
  for (int j = 0; j < NCOLS / (32 * 4); ++j) {
    const int e4 = j * 32 + lane;                       // float4 index
    const uint32_t lds_off = (uint32_t)(size_t)(&sy[e4 * 4]);
    const float* g = src + e4 * 4;
    asm volatile("global_load_async_to_lds_b128 %0, %1, off"
                 :: "v"(lds_off), "v"(g)
                 : "memory");
  }
  asm volatile("s_wait_asynccnt 0x0" ::: "memory");
#else
  for (int j = 0; j < NCOLS / (32 * 4); ++j) {
    const int e4 = j * 32 + lane;
    ((float4*)sy)[e4] = ((const float4*)src)[e4];
  }
#endif
  __syncthreads();  // single-wave workgroup: barrier lowers to S_NOP

  // ---- Phase 2: Condat TV-1D (serial, lane 0), faithful port of reference ----
  if (lane == 0) {
    const float lam = LAMBDA;
    int k = 0, k0 = 0, km = 0, kp = 0;
    float vmin = sy[0] - lam, vmax = sy[0] + lam;
    float umin = lam, umax = -lam;
    bool done = false;
    int guard = 0;
    while (!done && (guard++ < (1 << 20))) {
      const bool at_end = (k == NCOLS - 1);
      const float ynext = sy[at_end ? (NCOLS - 1) : (k + 1)];
      const bool neg = at_end ? (umin < 0.0f) : (ynext + umin < vmin - lam);
      const bool pos = (!neg) && (at_end ? (umax > 0.0f)
                                         : (ynext + umax > vmax + lam));
      const bool mid = (!neg) && (!pos);
      const bool fin = mid && at_end;
      if (neg || pos || fin) {                 // flush a finished segment
        const int hi = neg ? km : (pos ? kp : (NCOLS - 1));
        const float cnt_fin = (float)(k - k0 + 1);
        const float val = pos ? vmax
                              : (fin ? (vmin + umin / cnt_fin) : vmin);
        for (int i = k0; i <= hi; ++i) sx[i] = val;
      }
      int nk = neg ? (km + 1) : (pos ? (kp + 1) : (k + 1));
      if (nk > NCOLS - 1) nk = NCOLS - 1;
      const float ynk = sy[nk];
      const float umin_t = umin + ynk - vmin;
      const float umax_t = umax + ynk - vmax;
      const float cnt = (float)(nk - k0 + 1);
      const bool c1 = (umin_t >= lam);
      const bool c2 = (umax_t <= -lam);
      const float vmin_mid = c1 ? (vmin + (umin_t - lam) / cnt) : vmin;
      const float umin_mid = c1 ? lam : umin_t;
      const int   km_mid   = c1 ? nk : km;
      const float vmax_mid = c2 ? (vmax + (umax_t + lam) / cnt) : vmax;
      const float umax_mid = c2 ? -lam : umax_t;
      const int   kp_mid   = c2 ? nk : kp;
      const bool mneg = neg && !at_end, mpos = pos && !at_end;
      const bool mmid = mid && !at_end;
      const bool eneg = neg && at_end,  epos = pos && at_end;
      const bool jump = neg || pos;
      const int nk0 = jump ? nk : k0;
      const int nkm = (neg || mpos) ? nk : (mmid ? km_mid : km);
      const int nkp = (pos || mneg) ? nk : (mmid ? kp_mid : kp);
      const float nvmin = neg ? ynk
                        : (mpos ? (ynk - 2.0f * lam)
                        : (mmid ? vmin_mid : vmin));
      const float nvmax = pos ? ynk
                        : (mneg ? (ynk + 2.0f * lam)
                        : (mmid ? vmax_mid : vmax));
      const float numin = (neg || mpos) ? lam
                        : (epos ? (ynk - lam - vmin)
                        : (mmid ? umin_mid : umin));
      const float numax = (pos || mneg) ? -lam
                        : (eneg ? (ynk + lam - vmax)
                        : (mmid ? umax_mid : umax));
      k = fin ? k : nk;
      k0 = nk0; km = nkm; kp = nkp;
      vmin = nvmin; vmax = nvmax; umin = numin; umax = numax;
      done = done || fin;
    }
  }
  __syncthreads();

  // ---- Phase 3: sparsemax via Michelot (exact simplex projection) ----
  float v[16];
  float s = 0.0f;
  #pragma unroll
  for (int j = 0; j < 16; ++j) {
    v[j] = sx[lane + 32 * j];     // conflict-free: distinct banks per lane
    s += v[j];
  }
  #pragma unroll
  for (int off = 16; off > 0; off >>= 1) s += __shfl_xor(s, off, 32);

  int cnt = NCOLS;
  float tau = (s - 1.0f) / (float)NCOLS;
  for (int it = 0; it < NCOLS; ++it) {   // support strictly shrinks -> terminates
    float ls = 0.0f;
    int lc = 0;
    #pragma unroll
    for (int j = 0; j < 16; ++j) {
      if (v[j] > tau) { ls += v[j]; ++lc; }
    }
    #pragma unroll
    for (int off = 16; off > 0; off >>= 1) {
      ls += __shfl_xor(ls, off, 32);
      lc += __shfl_xor(lc, off, 32);
    }
    if (lc == cnt || lc == 0) break;
    cnt = lc;
    tau = (ls - 1.0f) / (float)lc;
  }

  float* __restrict__ dst = out + (size_t)row * NCOLS;
  #pragma unroll
  for (int j = 0; j < 16; ++j) {
    const float d = v[j] - tau;
    dst[lane + 32 * j] = d > 0.0f ? d : 0.0f;
  }
}

extern "C" void kernel_launch(void* const* d_in, const int* in_sizes, int n_in,
                              void* d_out, int out_size, void* d_ws, size_t ws_size,
                              hipStream_t stream) {
  (void)n_in; (void)out_size; (void)d_ws; (void)ws_size;
  const float* x = (const float*)d_in[0];
  float* out = (float*)d_out;
  const int nrows = in_sizes[0] / NCOLS;   // B = 4096 for the reference shapes
  fusedmax_tv_sparsemax<<<nrows, 32, 0, stream>>>(x, out, nrows);
}